// mySoftmax_77635828842932
// MI455X (gfx1250) — compile-verified
//
#include <hip/hip_runtime.h>

// ---------------------------------------------------------------------------
// Segmented tiny-row softmax, bandwidth-bound (~794 MB traffic, ~34us floor
// at 23.3 TB/s on MI455X). Per-workgroup 84KB LDS tile of 64 macro-blocks
// staged via gfx1250 async global<->LDS DMA (ASYNCcnt) with immediate
// offsets (applied to both LDS and global sides per ISA), NT cache policy
// for the one-pass stream, per-thread softmax over width-8/11 rows in LDS.
// Reference shape: 294912 macro-blocks = 4608 full tiles (exact division).
// ---------------------------------------------------------------------------

#define BLOCK_F     336                   // floats per macro-block
#define WG_MB       64                    // macro-blocks per workgroup
#define TILE_F      (WG_MB * BLOCK_F)     // 21504 floats = 84 KB LDS
#define TILE_V4     (TILE_F / 4)          // 5376 float4
#define THREADS     256                   // 8 waves (wave32)
#define V4_PER_THR  (TILE_V4 / THREADS)   // 21 DMA ops per thread per phase
#define STEP_B      (THREADS * 16)        // 4096-byte stride between DMA ops
#define ROWS_PER_MB 33
#define TILE_ROWS   (WG_MB * ROWS_PER_MB) // 2112 softmax rows per tile
#define CPOL_NT     1                     // TH[2:0]=1 -> non-temporal

typedef int v4i __attribute__((vector_size(16)));             // int4 payload
typedef __attribute__((address_space(1))) v4i* g4ptr_t;       // global int4*
typedef __attribute__((address_space(3))) v4i* l4ptr_t;       // LDS int4*

__device__ __forceinline__ void wait_async0() {
#if __has_builtin(__builtin_amdgcn_s_wait_asynccnt)
    __builtin_amdgcn_s_wait_asynccnt(0);
#else
    asm volatile("s_wait_asynccnt 0" ::: "memory");
#endif
}

// Template-recursive unroll so the DMA instruction offset is a frontend
// integer constant expression at every call site.
template <int K>
struct AsyncLoads {
    __device__ static __forceinline__ void run(g4ptr_t g, l4ptr_t l) {
        AsyncLoads<K - 1>::run(g, l);
        __builtin_amdgcn_global_load_async_to_lds_b128(g, l, (K - 1) * STEP_B,
                                                       CPOL_NT);
    }
};
template <>
struct AsyncLoads<0> {
    __device__ static __forceinline__ void run(g4ptr_t, l4ptr_t) {}
};

template <int K>
struct AsyncStores {
    __device__ static __forceinline__ void run(g4ptr_t g, l4ptr_t l) {
        AsyncStores<K - 1>::run(g, l);
        __builtin_amdgcn_global_store_async_from_lds_b128(g, l, (K - 1) * STEP_B,
                                                          CPOL_NT);
    }
};
template <>
struct AsyncStores<0> {
    __device__ static __forceinline__ void run(g4ptr_t, l4ptr_t) {}
};

__global__ __launch_bounds__(THREADS)
void mySoftmax_77635828842932_kernel(const float* __restrict__ x,
                                     float* __restrict__ out) {
    __shared__ float tile[TILE_F];

    const int tid = threadIdx.x;
    const long long wg_base_f = (long long)blockIdx.x * TILE_F; // float index

    // One base address per thread; the 24-bit instruction offset advances
    // both the global and LDS sides of each async DMA (ISA: INST_OFFSET is
    // added to VADDR and VDST addresses alike).
    const g4ptr_t gsrc  = (g4ptr_t)(x   + wg_base_f + tid * 4);
    const g4ptr_t gdst  = (g4ptr_t)(out + wg_base_f + tid * 4);
    const l4ptr_t lbase = (l4ptr_t)&tile[tid * 4];

    // ---- Phase A: async DMA global -> LDS, 16B/lane, clause of 21 ops ------
    AsyncLoads<V4_PER_THR>::run(gsrc, lbase);
    wait_async0();
    __syncthreads();

    // ---- Phase B: per-thread softmax over width-8/11 rows in LDS -----------
    for (int row = tid; row < TILE_ROWS; row += THREADS) {
        const int mb = row / ROWS_PER_MB;
        const int lr = row - mb * ROWS_PER_MB;
        int off, c, rr;
        if (lr < 9)       { off = 0;   c = 11; rr = lr;      }
        else if (lr < 18) { off = 99;  c = 11; rr = lr - 9;  }
        else if (lr < 27) { off = 198; c = 8;  rr = lr - 18; }
        else              { off = 270; c = 11; rr = lr - 27; }

        float* p = &tile[mb * BLOCK_F + off + rr * c];

        float m = p[0];
        for (int i = 1; i < c; ++i) m = fmaxf(m, p[i]);
        float s = 0.0f;
        for (int i = 0; i < c; ++i) {
            const float e = __expf(p[i] - m);
            p[i] = e;
            s += e;
        }
        const float inv = 1.0f / s;
        for (int i = 0; i < c; ++i) p[i] *= inv;
    }
    __syncthreads();

    // ---- Phase C: async DMA LDS -> global, clause of 21 ops ----------------
    AsyncStores<V4_PER_THR>::run(gdst, lbase);
    wait_async0();
}

extern "C" void kernel_launch(void* const* d_in, const int* in_sizes, int n_in,
                              void* d_out, int out_size, void* d_ws, size_t ws_size,
                              hipStream_t stream) {
    (void)n_in; (void)d_ws; (void)ws_size; (void)out_size;
    const float* x = (const float*)d_in[0];
    float* out = (float*)d_out;

    const long long total_f = (long long)in_sizes[0];   // 32768*3024
    const int n_mb = (int)(total_f / BLOCK_F);          // 294912
    const int grid = n_mb / WG_MB;                      // 4608 full tiles (exact)

    mySoftmax_77635828842932_kernel<<<grid, THREADS, 0, stream>>>(x, out);
}